// ResidualVectorQuantizer_21277267985088
// MI455X (gfx1250) — compile-verified
//
#include <hip/hip_runtime.h>
#include <hip/hip_bf16.h>

typedef float v2f __attribute__((ext_vector_type(2)));
typedef float v8f __attribute__((ext_vector_type(8)));

#define DIM          128
#define NCODES       1024
#define NSTAGES      4
#define CHUNK        64                 // codes staged in LDS per pass
#define NCHUNKS      (NCODES / CHUNK)   // 16
#define TSTRIDE      132                // padded LDS stride (floats): 132 % 64 = 4 -> conflict-free b64 reads
#define WAVES        8
#define ROWS_PER_WAVE 16
#define ROWS_PER_BLOCK (WAVES * ROWS_PER_WAVE)  // 128

__global__ __launch_bounds__(256) void rvq_wmma_kernel(
    const float* __restrict__ x,        // [N, 128]
    const float* __restrict__ cb,       // [4, 1024, 128]
    float* __restrict__ q_out,          // [N, 128]
    float* __restrict__ idx_out,        // [4, N] (indices stored as float)
    int nrows)
{
    __shared__ float tile[CHUNK * TSTRIDE];          // ~33.8 KB
    __shared__ float norms[CHUNK];
    __shared__ int   bestLds[WAVES][ROWS_PER_WAVE];

    const int tid  = threadIdx.x;
    const int wave = tid >> 5;
    const int lane = tid & 31;
    const int col  = lane & 15;   // A: row-in-tile; B/D: column
    const int hi   = lane >> 4;   // K sub-offset selector (A/B), +8 row offset (C/D)

    const int rowBase = blockIdx.x * ROWS_PER_BLOCK + wave * ROWS_PER_WAVE;
    const int myRow   = rowBase + col;

    // ---- Load residual tile as WMMA A-fragments (ISA 32-bit A 16x4 layout):
    // lane(col,hi), step kb holds residual[myRow][4*kb + 2*hi + {0,1}]
    v2f a[DIM / 4];
    const float* xrow = x + (size_t)myRow * DIM;
    #pragma unroll
    for (int kb = 0; kb < DIM / 4; ++kb)
        a[kb] = *(const v2f*)(xrow + kb * 4 + 2 * hi);

    for (int s = 0; s < NSTAGES; ++s) {
        const float* cbs = cb + (size_t)s * NCODES * DIM;

        float bestD[8];
        int   bestI[8];
        #pragma unroll
        for (int v = 0; v < 8; ++v) { bestD[v] = 3.4e38f; bestI[v] = 0x7fffffff; }

        for (int ch = 0; ch < NCHUNKS; ++ch) {
            __syncthreads();
            // ---- cooperative codebook chunk load: 64 codes x 128 floats + norms
            {
                const int code    = tid >> 2;     // 0..63
                const int quarter = tid & 3;      // 32 floats each
                const float* src = cbs + (size_t)(ch * CHUNK + code) * DIM + quarter * 32;
                float*       dst = tile + code * TSTRIDE + quarter * 32;
                float psum = 0.f;
                #pragma unroll
                for (int j = 0; j < 8; ++j) {
                    float4 vv = *(const float4*)(src + j * 4);
                    *(float4*)(dst + j * 4) = vv;
                    psum += vv.x * vv.x + vv.y * vv.y + vv.z * vv.z + vv.w * vv.w;
                }
                psum += __shfl_xor(psum, 1, 32);
                psum += __shfl_xor(psum, 2, 32);
                if (quarter == 0) norms[code] = psum;
            }
            if (ch + 1 < NCHUNKS) {   // warm L2/WGP$ for the next chunk
                __builtin_prefetch(cbs + (size_t)((ch + 1) * CHUNK) * DIM + tid * 32, 0, 0);
            }
            __syncthreads();

            // ---- 2 pairs of N-tiles; each pair = two independent WMMA chains
            //      interleaved so the matrix pipe always has a ready op.
            #pragma unroll
            for (int nt = 0; nt < CHUNK / 16; nt += 2) {
                v8f acc0 = {0.f, 0.f, 0.f, 0.f, 0.f, 0.f, 0.f, 0.f};
                v8f acc1 = {0.f, 0.f, 0.f, 0.f, 0.f, 0.f, 0.f, 0.f};
                const float* bbase0 = tile + ((nt + 0) * 16 + col) * TSTRIDE + 2 * hi;
                const float* bbase1 = tile + ((nt + 1) * 16 + col) * TSTRIDE + 2 * hi;
                #pragma unroll
                for (int kb = 0; kb < DIM / 4; ++kb) {
                    v2f b0 = *(const v2f*)(bbase0 + kb * 4);   // ds_load_b64, conflict-free
                    v2f b1 = *(const v2f*)(bbase1 + kb * 4);
                    acc0 = __builtin_amdgcn_wmma_f32_16x16x4_f32(
                        false, a[kb], false, b0, (short)0, acc0, false, false);
                    acc1 = __builtin_amdgcn_wmma_f32_16x16x4_f32(
                        false, a[kb], false, b1, (short)0, acc1, false, false);
                }
                const float cn0 = norms[(nt + 0) * 16 + col];
                const float cn1 = norms[(nt + 1) * 16 + col];
                const int   ci0 = ch * CHUNK + (nt + 0) * 16 + col;
                const int   ci1 = ch * CHUNK + (nt + 1) * 16 + col;
                #pragma unroll
                for (int v = 0; v < 8; ++v) {
                    float d0 = cn0 - 2.0f * acc0[v];   // ||r||^2 row-constant: dropped
                    if (d0 < bestD[v]) { bestD[v] = d0; bestI[v] = ci0; }
                    float d1 = cn1 - 2.0f * acc1[v];
                    if (d1 < bestD[v]) { bestD[v] = d1; bestI[v] = ci1; }
                }
            }
        }

        // ---- per-row argmin across the 16 lanes of each half (rows v + 8*hi)
        #pragma unroll
        for (int m = 1; m < 16; m <<= 1) {
            #pragma unroll
            for (int v = 0; v < 8; ++v) {
                float od = __shfl_xor(bestD[v], m, 32);
                int   oi = __shfl_xor(bestI[v], m, 32);
                if (od < bestD[v] || (od == bestD[v] && oi < bestI[v])) {
                    bestD[v] = od; bestI[v] = oi;
                }
            }
        }
        if (col == 0) {
            #pragma unroll
            for (int v = 0; v < 8; ++v) bestLds[wave][v + 8 * hi] = bestI[v];
        }
        __syncthreads();

        const int myBest = bestLds[wave][col];   // argmin for row myRow
        if (hi == 0)
            idx_out[(size_t)s * nrows + myRow] = (float)myBest;

        // ---- residual -= chosen code (same fragment addressing; L2-hot)
        const float* crow = cbs + (size_t)myBest * DIM;
        #pragma unroll
        for (int kb = 0; kb < DIM / 4; ++kb) {
            v2f c2 = *(const v2f*)(crow + kb * 4 + 2 * hi);
            a[kb] -= c2;
        }
    }

    // ---- quantized = x - residual_final
    float* qrow = q_out + (size_t)myRow * DIM;
    #pragma unroll
    for (int kb = 0; kb < DIM / 4; ++kb) {
        v2f xv = *(const v2f*)(xrow + kb * 4 + 2 * hi);
        *(v2f*)(qrow + kb * 4 + 2 * hi) = xv - a[kb];
    }
}

extern "C" void kernel_launch(void* const* d_in, const int* in_sizes, int n_in,
                              void* d_out, int out_size, void* d_ws, size_t ws_size,
                              hipStream_t stream) {
    const float* x  = (const float*)d_in[0];   // [16, 8192, 128] f32
    const float* cb = (const float*)d_in[1];   // [4, 1024, 128] f32
    const int nrows = in_sizes[0] / DIM;       // 131072

    float* q   = (float*)d_out;                // first output: quantized [N,128]
    float* idx = q + (size_t)nrows * DIM;      // second output: indices [4,N] as f32

    dim3 grid(nrows / ROWS_PER_BLOCK);         // 1024 blocks
    dim3 block(WAVES * 32);                    // 8 wave32s
    rvq_wmma_kernel<<<grid, block, 0, stream>>>(x, cb, q, idx, nrows);
}